// CML_Model_48859547959346
// MI455X (gfx1250) — compile-verified
//
#include <hip/hip_runtime.h>
#include <hip/hip_bf16.h>

#define HID 8
#define NA 11
#define NE 22
#define NTOK 242          // NE * NA
#define NLAYER 6
#define H2 16
#define QNUM 22
#define NROWS (1 << QNUM)
#define BOS_IDX 0x2AAAAA  // one-hot index of the HF product state
#define BOS_AMP 2048.0f   // 2^(QNUM/2)

typedef float v2f __attribute__((ext_vector_type(2)));
typedef float v8f __attribute__((ext_vector_type(8)));

// ---------------------------------------------------------------------------
// Kernel 1: the entire tiny network in one workgroup -> 16-float vector in ws
// ---------------------------------------------------------------------------
__global__ __launch_bounds__(256) void small_net_kernel(
    const float* __restrict__ pos_a, const int* __restrict__ ix_a,
    const int* __restrict__ pos_ix, const int* __restrict__ atom_ix,
    const float* __restrict__ rpos_w, const float* __restrict__ emb_w,
    const float* __restrict__ emb_b,
    const float* __restrict__ Wq, const float* __restrict__ bq,
    const float* __restrict__ Wk, const float* __restrict__ bk,
    const float* __restrict__ Wv, const float* __restrict__ bv,
    const float* __restrict__ Wo, const float* __restrict__ bo,
    const float* __restrict__ W1, const float* __restrict__ b1,
    const float* __restrict__ W2, const float* __restrict__ b2,
    const float* __restrict__ ln1_g, const float* __restrict__ ln1_b,
    const float* __restrict__ ln2_g, const float* __restrict__ ln2_b,
    const float* __restrict__ Wi, const float* __restrict__ bi,
    const float* __restrict__ ni_g, const float* __restrict__ ni_b,
    const float* __restrict__ conv_a_w, const float* __restrict__ conv_e_w,
    float* __restrict__ xout) {
  __shared__ float s_seq[NTOK * HID];   // 1936
  __shared__ float s_x[NTOK * HID];     // 1936
  __shared__ float s_un[8448];          // union: q/k/v (3*1936) OR conv bufA/bufB (2*4224)
  __shared__ float s_rae[NTOK];
  __shared__ float s_r[NTOK];
  __shared__ float s_ampr[NE];
  __shared__ float s_y[NE * H2];
  __shared__ float s_xe[H2 * NE];       // [c][e]
  __shared__ float s_eb[2 * H2 * 24];   // conv_e double buffer
  __shared__ float s_scal[8];           // 0:amp_ae 1:bias_ae 2:r_mean 3:r_std 4:amp_r2
  __shared__ float s_aeinv[HID];
  __shared__ float s_y2[H2];

  const int t = threadIdx.x;
  const float inv_scale = 1.0f / sqrtf((float)HID);

  // ---- embedding: pos_e, ae, r_ae, seq ----
  if (t < NTOK) {
    const int e = t / NA, a = t % NA;
    float aev[3], rr = 0.0f;
    for (int j = 0; j < 3; ++j) {
      float pe = rpos_w[pos_ix[e] * 3 + j] + pos_a[atom_ix[e] * 3 + j];
      aev[j] = pe - pos_a[a * 3 + j];
      rr += aev[j] * aev[j];
    }
    rr = sqrtf(rr);
    s_rae[t] = rr;
    for (int h = 0; h < HID; ++h) {
      float s = emb_b[h] + rr * emb_w[h * 4 + 3];
      for (int j = 0; j < 3; ++j) s += aev[j] * emb_w[h * 4 + j];
      s_seq[t * HID + h] = s;
    }
  }
  __syncthreads();

  if (t == 0) {  // amp_ae = std(r_ae, ddof=1), bias_ae = mean(r_ae)
    float m = 0.0f;
    for (int i = 0; i < NTOK; ++i) m += s_rae[i];
    m /= (float)NTOK;
    float ss = 0.0f;
    for (int i = 0; i < NTOK; ++i) { float d = s_rae[i] - m; ss += d * d; }
    s_scal[0] = sqrtf(ss / (float)(NTOK - 1));
    s_scal[1] = m;
  }
  __syncthreads();

  // ---- transformer layers ----
  float* sq = s_un;
  float* sk = s_un + NTOK * HID;
  float* sv = s_un + 2 * NTOK * HID;
  for (int l = 0; l < NLAYER; ++l) {
    if (t < NTOK) {
      const int a = t % NA;
      const float amp = (float)ix_a[a];
      float xv[HID];
      for (int h = 0; h < HID; ++h) { xv[h] = amp * s_seq[t * HID + h]; s_x[t * HID + h] = xv[h]; }
      for (int h = 0; h < HID; ++h) {
        float qh = bq[l * HID + h], kh = bk[l * HID + h], vh = bv[l * HID + h];
        for (int h2 = 0; h2 < HID; ++h2) {
          qh += xv[h2] * Wq[(l * HID + h) * HID + h2];
          kh += xv[h2] * Wk[(l * HID + h) * HID + h2];
          vh += xv[h2] * Wv[(l * HID + h) * HID + h2];
        }
        sq[t * HID + h] = qh; sk[t * HID + h] = kh; sv[t * HID + h] = vh;
      }
    }
    __syncthreads();
    if (t < NTOK) {
      const int e = t / NA;
      float sc[NA], mx = -1e30f;
      for (int ka = 0; ka < NA; ++ka) {
        float s = 0.0f;
        for (int h = 0; h < HID; ++h) s += sq[t * HID + h] * sk[(e * NA + ka) * HID + h];
        s *= inv_scale;
        sc[ka] = s; mx = fmaxf(mx, s);
      }
      float ssum = 0.0f;
      for (int ka = 0; ka < NA; ++ka) { sc[ka] = expf(sc[ka] - mx); ssum += sc[ka]; }
      const float inv_sum = 1.0f / ssum;
      float attv[HID];
      for (int h = 0; h < HID; ++h) {
        float av = 0.0f;
        for (int ka = 0; ka < NA; ++ka) av += sc[ka] * sv[(e * NA + ka) * HID + h];
        attv[h] = av * inv_sum;
      }
      float x1[HID];
      for (int h = 0; h < HID; ++h) {
        float ao = bo[l * HID + h];
        for (int h2 = 0; h2 < HID; ++h2) ao += attv[h2] * Wo[(l * HID + h) * HID + h2];
        x1[h] = s_x[t * HID + h] + ao;
      }
      {  // LN1
        float m = 0.0f; for (int h = 0; h < HID; ++h) m += x1[h];
        m /= (float)HID;
        float v = 0.0f; for (int h = 0; h < HID; ++h) { float d = x1[h] - m; v += d * d; }
        v /= (float)HID;
        const float is = 1.0f / sqrtf(v + 1e-5f);
        for (int h = 0; h < HID; ++h)
          x1[h] = (x1[h] - m) * is * ln1_g[l * HID + h] + ln1_b[l * HID + h];
      }
      float f1[4 * HID];
      for (int o = 0; o < 4 * HID; ++o) {
        float s = b1[l * 4 * HID + o];
        for (int h = 0; h < HID; ++h) s += x1[h] * W1[(l * 4 * HID + o) * HID + h];
        f1[o] = fmaxf(s, 0.0f);
      }
      float x2[HID];
      for (int h = 0; h < HID; ++h) {
        float s = b2[l * HID + h];
        for (int o = 0; o < 4 * HID; ++o) s += f1[o] * W2[(l * HID + h) * 4 * HID + o];
        x2[h] = x1[h] + s;
      }
      {  // LN2
        float m = 0.0f; for (int h = 0; h < HID; ++h) m += x2[h];
        m /= (float)HID;
        float v = 0.0f; for (int h = 0; h < HID; ++h) { float d = x2[h] - m; v += d * d; }
        v /= (float)HID;
        const float is = 1.0f / sqrtf(v + 1e-5f);
        for (int h = 0; h < HID; ++h)
          x2[h] = (x2[h] - m) * is * ln2_g[l * HID + h] + ln2_b[l * HID + h];
      }
      for (int h = 0; h < HID; ++h) s_seq[t * HID + h] = x2[h];
    }
    __syncthreads();
  }

  // ---- ae_inv row 3 via 4x4 Gauss-Jordan on thread 0 ----
  if (t == 0) {
    float M[4][8];
    for (int i = 0; i < 4; ++i)
      for (int j = 0; j < 4; ++j) {
        float g = 0.0f;
        for (int h = 0; h < HID; ++h) g += emb_w[h * 4 + i] * emb_w[h * 4 + j];
        M[i][j] = g;
        M[i][4 + j] = (i == j) ? 1.0f : 0.0f;
      }
    for (int col = 0; col < 4; ++col) {
      int piv = col;
      for (int r2 = col + 1; r2 < 4; ++r2)
        if (fabsf(M[r2][col]) > fabsf(M[piv][col])) piv = r2;
      if (piv != col)
        for (int j = 0; j < 8; ++j) { float tmp = M[col][j]; M[col][j] = M[piv][j]; M[piv][j] = tmp; }
      const float inv = 1.0f / M[col][col];
      for (int j = 0; j < 8; ++j) M[col][j] *= inv;
      for (int r2 = 0; r2 < 4; ++r2) {
        if (r2 == col) continue;
        const float f = M[r2][col];
        for (int j = 0; j < 8; ++j) M[r2][j] -= f * M[col][j];
      }
    }
    for (int h = 0; h < HID; ++h) {
      float s = 0.0f;
      for (int j = 0; j < 4; ++j) s += M[3][4 + j] * emb_w[h * 4 + j];
      s_aeinv[h] = s;
    }
  }
  __syncthreads();

  // ---- r projection + normalization ----
  if (t < NTOK) {
    float s = 0.0f;
    for (int h = 0; h < HID; ++h) s += s_aeinv[h] * s_seq[t * HID + h];
    s_r[t] = s;
  }
  __syncthreads();
  if (t == 0) {
    float m = 0.0f;
    for (int i = 0; i < NTOK; ++i) m += s_r[i];
    m /= (float)NTOK;
    float ss = 0.0f;
    for (int i = 0; i < NTOK; ++i) { float d = s_r[i] - m; ss += d * d; }
    s_scal[2] = m;
    s_scal[3] = sqrtf(ss / (float)(NTOK - 1));
  }
  __syncthreads();
  if (t < NTOK) {
    float rr = s_scal[0] * (s_r[t] - s_scal[2]) / s_scal[3] + s_scal[1];
    s_r[t] = expf(-rr);  // keep exp(-r)
  }
  __syncthreads();
  if (t < NE) {
    float s = 0.0f;
    for (int a = 0; a < NA; ++a) s += s_r[t * NA + a];
    s_ampr[t] = s / (float)NA;
  }

  // ---- x = (exp(-r)*amp*seq) @ Wi.T + bi, transposed to [e][c][a]; y mean ----
  float* bufA = s_un;
  float* bufB = s_un + 4224;
  for (int idx = t; idx < NE * H2; idx += 256) {
    const int e = idx / H2, c = idx % H2;
    float ysum = 0.0f;
    for (int a = 0; a < NA; ++a) {
      const float amp = (float)ix_a[a];
      float dot = 0.0f;
      for (int h = 0; h < HID; ++h) dot += s_seq[(e * NA + a) * HID + h] * Wi[c * HID + h];
      const float val = s_r[e * NA + a] * amp * dot + bi[c];
      bufA[(e * H2 + c) * 12 + a] = val;
      ysum += val;
    }
    s_y[e * H2 + c] = ysum / (float)NA;
  }
  __syncthreads();

  // ---- conv_a chain: 6 iterations of zero-pad + stride-2 conv ----
  {
    float* cur = bufA; float* nxt = bufB;
    int len = NA;
    for (int it = 0; it < 6; ++it) {
      const int nlen = (len - 1) / 2 + 1;
      for (int idx = t; idx < NE * H2; idx += 256) {
        const int e = idx / H2, o = idx % H2;
        for (int p = 0; p < nlen; ++p) {
          float s = 0.0f;
          for (int i = 0; i < H2; ++i)
            for (int kk = 0; kk < 2; ++kk) {
              const int pos = 2 * p + kk;
              if (pos < len) s += cur[(e * H2 + i) * 12 + pos] * conv_a_w[(o * H2 + i) * 2 + kk];
            }
          nxt[(e * H2 + o) * 12 + p] = s;
        }
      }
      __syncthreads();
      float* tmp = cur; cur = nxt; nxt = tmp;
      len = nlen;
    }
    // ---- per-electron LN, scale by amp_r, transpose to [c][e] ----
    if (t < NE) {
      float z[H2]; float m = 0.0f;
      for (int c = 0; c < H2; ++c) { z[c] = s_y[t * H2 + c] + cur[(t * H2 + c) * 12 + 0]; m += z[c]; }
      m /= (float)H2;
      float v = 0.0f;
      for (int c = 0; c < H2; ++c) { float d = z[c] - m; v += d * d; }
      v /= (float)H2;
      const float is = 1.0f / sqrtf(v + 1e-5f);
      const float ar = s_ampr[t];
      for (int c = 0; c < H2; ++c) s_xe[c * NE + t] = ar * ((z[c] - m) * is * ni_g[c] + ni_b[c]);
    }
  }
  __syncthreads();
  if (t < H2) {
    float s = 0.0f;
    for (int e = 0; e < NE; ++e) s += s_xe[t * NE + e];
    s_y2[t] = s / (float)NE;
    for (int e = 0; e < NE; ++e) s_eb[t * 24 + e] = s_xe[t * NE + e];
  }
  if (t == 0) {
    float s = 0.0f;
    for (int e = 0; e < NE; ++e) s += s_ampr[e];
    s_scal[4] = s / (float)NE;
  }
  __syncthreads();

  // ---- conv_e chain: 11 iterations ----
  {
    float* ec = s_eb; float* en = s_eb + H2 * 24;
    int len = NE;
    for (int it = 0; it < 11; ++it) {
      const int nlen = (len - 1) / 2 + 1;
      if (t < H2) {
        const int o = t;
        for (int p = 0; p < nlen; ++p) {
          float s = 0.0f;
          for (int i = 0; i < H2; ++i)
            for (int kk = 0; kk < 2; ++kk) {
              const int pos = 2 * p + kk;
              if (pos < len) s += ec[i * 24 + pos] * conv_e_w[(o * H2 + i) * 2 + kk];
            }
          en[o * 24 + p] = s;
        }
      }
      __syncthreads();
      float* tmp = ec; ec = en; en = tmp;
      len = nlen;
    }
    // ---- final LN + scale -> xout[16] ----
    if (t < H2) {
      float m = 0.0f;
      for (int c = 0; c < H2; ++c) m += s_y2[c] + ec[c * 24 + 0];
      m /= (float)H2;
      float v = 0.0f;
      for (int c = 0; c < H2; ++c) { float d = s_y2[c] + ec[c * 24 + 0] - m; v += d * d; }
      v /= (float)H2;
      const float is = 1.0f / sqrtf(v + 1e-5f);
      const float z = s_y2[t] + ec[t * 24 + 0];
      xout[t] = s_scal[4] * ((z - m) * is * ni_g[t] + ni_b[t]);
    }
  }
}

// ---------------------------------------------------------------------------
// Kernel 2: psi = Wout @ x + bout + bos, via V_WMMA_F32_16X16X4_F32
//   D(16x16) = A(16x4, x broadcast per row) x B(4x16, Wout tile transposed)
//   Every D column is identical -> lanes 0..15 VGPR0 = psi[base+lane]
//   Wout (268 MB) is a read-once stream > 192 MB L2 -> non-temporal loads
//   so bout/out (16 MB each) stay L2-resident across graph replays.
// ---------------------------------------------------------------------------
__global__ __launch_bounds__(256) void gemv_wmma_kernel(
    const float* __restrict__ W, const float* __restrict__ bout,
    const float* __restrict__ xs, float* __restrict__ out, int nrows) {
  const int lane = threadIdx.x & 31;
  const int wave = (int)((blockIdx.x * blockDim.x + threadIdx.x) >> 5);
  const int nwaves = (int)((gridDim.x * blockDim.x) >> 5);
  const int hi = lane >> 4;    // 0: lanes 0-15 (K=0,1 / rows N=0..15), 1: lanes 16-31 (K=2,3)
  const int ln = lane & 15;

  // A operand: x chunk broadcast across all 16 rows (documented 16x4 f32 layout)
  v2f a0 = { xs[0 + 2 * hi],  xs[1 + 2 * hi]  };
  v2f a1 = { xs[4 + 2 * hi],  xs[5 + 2 * hi]  };
  v2f a2 = { xs[8 + 2 * hi],  xs[9 + 2 * hi]  };
  v2f a3 = { xs[12 + 2 * hi], xs[13 + 2 * hi] };

  const int ntiles = nrows >> 4;
  for (int tile = wave; tile < ntiles; tile += nwaves) {
    const int base = tile << 4;
    const float* rowp = W + (size_t)(base + ln) * 16 + 2 * hi;
    // prefetch this wave's next tile
    __builtin_prefetch(rowp + (size_t)nwaves * 256, 0, 1);
    v2f b0 = __builtin_nontemporal_load((const v2f*)(rowp + 0));
    v2f b1 = __builtin_nontemporal_load((const v2f*)(rowp + 4));
    v2f b2 = __builtin_nontemporal_load((const v2f*)(rowp + 8));
    v2f b3 = __builtin_nontemporal_load((const v2f*)(rowp + 12));
    v8f c = {};
    c = __builtin_amdgcn_wmma_f32_16x16x4_f32(false, a0, false, b0, (short)0, c, false, false);
    c = __builtin_amdgcn_wmma_f32_16x16x4_f32(false, a1, false, b1, (short)0, c, false, false);
    c = __builtin_amdgcn_wmma_f32_16x16x4_f32(false, a2, false, b2, (short)0, c, false, false);
    c = __builtin_amdgcn_wmma_f32_16x16x4_f32(false, a3, false, b3, (short)0, c, false, false);
    if (hi == 0) {
      const int idx = base + ln;
      float v = c[0] + bout[idx];
      if (idx == BOS_IDX) v += BOS_AMP;
      out[idx] = v;
    }
  }
}

extern "C" void kernel_launch(void* const* d_in, const int* in_sizes, int n_in,
                              void* d_out, int out_size, void* d_ws, size_t ws_size,
                              hipStream_t stream) {
  const float* pos_a  = (const float*)d_in[0];
  const int*   ix_a   = (const int*)d_in[1];
  const int*   pos_ix = (const int*)d_in[2];
  const int*   atom_ix= (const int*)d_in[3];
  const float* rpos_w = (const float*)d_in[4];
  const float* emb_w  = (const float*)d_in[5];
  const float* emb_b  = (const float*)d_in[6];
  const float* Wq = (const float*)d_in[7];  const float* bq = (const float*)d_in[8];
  const float* Wk = (const float*)d_in[9];  const float* bk = (const float*)d_in[10];
  const float* Wv = (const float*)d_in[11]; const float* bv = (const float*)d_in[12];
  const float* Wo = (const float*)d_in[13]; const float* bo = (const float*)d_in[14];
  const float* W1 = (const float*)d_in[15]; const float* b1 = (const float*)d_in[16];
  const float* W2 = (const float*)d_in[17]; const float* b2 = (const float*)d_in[18];
  const float* ln1_g = (const float*)d_in[19]; const float* ln1_b = (const float*)d_in[20];
  const float* ln2_g = (const float*)d_in[21]; const float* ln2_b = (const float*)d_in[22];
  const float* Wi = (const float*)d_in[23]; const float* bi = (const float*)d_in[24];
  const float* ni_g = (const float*)d_in[25]; const float* ni_b = (const float*)d_in[26];
  const float* conv_a_w = (const float*)d_in[27];
  const float* conv_e_w = (const float*)d_in[28];
  const float* Wout = (const float*)d_in[29];
  const float* bout = (const float*)d_in[30];

  float* xs = (float*)d_ws;  // 16 floats

  small_net_kernel<<<1, 256, 0, stream>>>(
      pos_a, ix_a, pos_ix, atom_ix, rpos_w, emb_w, emb_b,
      Wq, bq, Wk, bk, Wv, bv, Wo, bo, W1, b1, W2, b2,
      ln1_g, ln1_b, ln2_g, ln2_b, Wi, bi, ni_g, ni_b,
      conv_a_w, conv_e_w, xs);

  gemv_wmma_kernel<<<1024, 256, 0, stream>>>(Wout, bout, xs, (float*)d_out, NROWS);
}